// GCNModule_58342835749555
// MI455X (gfx1250) — compile-verified
//
#include <hip/hip_runtime.h>
#include <math.h>

typedef __attribute__((ext_vector_type(2))) float v2f;
typedef __attribute__((ext_vector_type(8))) float v8f;

#define DIM 64

// ---------------- zero workspace ----------------
__global__ void gcn_zero_kernel(float* __restrict__ p, int n) {
    int i = blockIdx.x * blockDim.x + threadIdx.x;
    int stride = gridDim.x * blockDim.x;
    for (; i < n; i += stride) p[i] = 0.0f;
}

// ---------------- degree accumulation ----------------
__global__ void gcn_degree_kernel(const int* __restrict__ src, const int* __restrict__ dst,
                                  float* __restrict__ out_deg, float* __restrict__ in_deg, int E) {
    int e = blockIdx.x * blockDim.x + threadIdx.x;
    if (e < E) {
        atomicAdd(&out_deg[src[e]], 1.0f);
        atomicAdd(&in_deg[dst[e]], 1.0f);
    }
}

// ---------------- per-edge norm ----------------
__global__ void gcn_norm_kernel(const int* __restrict__ src, const int* __restrict__ dst,
                                const float* __restrict__ out_deg, const float* __restrict__ in_deg,
                                float* __restrict__ normv, int E) {
    int e = blockIdx.x * blockDim.x + threadIdx.x;
    if (e < E) {
        float p = out_deg[src[e]] * in_deg[dst[e]];
        p = (p == 0.0f) ? 1.0f : p;
        normv[e] = 1.0f / sqrtf(p);
    }
}

// ---------------- scatter-add of x[src]*norm into agg[dst] ----------------
// 16 threads per edge, each handles a float4 (4 floats). All traffic L2-resident.
__global__ void gcn_scatter_kernel(const float* __restrict__ x, const int* __restrict__ src,
                                   const int* __restrict__ dst, const float* __restrict__ normv,
                                   float* __restrict__ agg, int E) {
    int idx = blockIdx.x * blockDim.x + threadIdx.x;
    int e = idx >> 4;
    if (e >= E) return;
    int c = (idx & 15) * 4;
    float nr = normv[e];
    const float4 v = *reinterpret_cast<const float4*>(x + (size_t)src[e] * DIM + c);
    float* ar = agg + (size_t)dst[e] * DIM + c;
    atomicAdd(ar + 0, v.x * nr);
    atomicAdd(ar + 1, v.y * nr);
    atomicAdd(ar + 2, v.z * nr);
    atomicAdd(ar + 3, v.w * nr);
}

// ---------------- FFN: (agg @ w1 + b1) -> GELU -> (@ w2 + b2), f32 WMMA ----------------
// One wave per 16-row tile; 4 waves per block. K=64 -> 16x V_WMMA_F32_16X16X4_F32 per
// 16x16 output tile, 4 N-tiles per GEMM, two GEMMs. GELU intermediate is staged through
// LDS to convert the C/D fragment layout back into A-fragment layout.
__global__ __launch_bounds__(128)
void gcn_ffn_wmma_kernel(const float* __restrict__ agg,
                         const float* __restrict__ w1, const float* __restrict__ b1,
                         const float* __restrict__ w2, const float* __restrict__ b2,
                         float* __restrict__ out, int nTiles) {
    __shared__ float hs[4][16][68];   // per-wave 16x64 GELU tile, stride 68 to spread banks

    const int lane = threadIdx.x & 31;
    const int wave = threadIdx.x >> 5;
    const int tile = blockIdx.x * 4 + wave;
    const bool active = (tile < nTiles);   // wave-uniform: EXEC stays all-1s inside

    const int row  = lane & 15;        // M row for A-frags / N col for B-frags
    const int half = lane >> 4;        // lane half selects K pair
    const int kk   = half * 2;
    const int base = tile * 16;

    v8f acc[4] = {};

    if (active) {
        // GEMM 1: h = agg_tile[16x64] @ w1[64x64]
        #pragma unroll
        for (int k0 = 0; k0 < DIM; k0 += 4) {
            const float* ap = agg + (size_t)(base + row) * DIM + (k0 + kk);
            v2f a; a.x = ap[0]; a.y = ap[1];
            #pragma unroll
            for (int nt = 0; nt < 4; ++nt) {
                const int col = nt * 16 + row;
                v2f b;
                b.x = w1[(k0 + kk)     * DIM + col];
                b.y = w1[(k0 + kk + 1) * DIM + col];
                acc[nt] = __builtin_amdgcn_wmma_f32_16x16x4_f32(
                    false, a, false, b, (short)0, acc[nt], false, false);
            }
        }
        // bias + exact GELU, stage to LDS in row-major tile form
        #pragma unroll
        for (int nt = 0; nt < 4; ++nt) {
            const float bias = b1[nt * 16 + row];
            #pragma unroll
            for (int r = 0; r < 8; ++r) {
                const int m = r + half * 8;
                float v = acc[nt][r] + bias;
                v = 0.5f * v * (1.0f + erff(v * 0.70710678118654752f));
                hs[wave][m][nt * 16 + row] = v;
            }
        }
    }
    __syncthreads();

    if (active) {
        // GEMM 2: out = h[16x64] @ w2[64x64] + b2
        v8f acc2[4] = {};
        #pragma unroll
        for (int k0 = 0; k0 < DIM; k0 += 4) {
            v2f a;
            a.x = hs[wave][row][k0 + kk];
            a.y = hs[wave][row][k0 + kk + 1];
            #pragma unroll
            for (int nt = 0; nt < 4; ++nt) {
                const int col = nt * 16 + row;
                v2f b;
                b.x = w2[(k0 + kk)     * DIM + col];
                b.y = w2[(k0 + kk + 1) * DIM + col];
                acc2[nt] = __builtin_amdgcn_wmma_f32_16x16x4_f32(
                    false, a, false, b, (short)0, acc2[nt], false, false);
            }
        }
        #pragma unroll
        for (int nt = 0; nt < 4; ++nt) {
            const float bias = b2[nt * 16 + row];
            #pragma unroll
            for (int r = 0; r < 8; ++r) {
                const int m = r + half * 8;
                out[(size_t)(base + m) * DIM + nt * 16 + row] = acc2[nt][r] + bias;
            }
        }
    }
}

extern "C" void kernel_launch(void* const* d_in, const int* in_sizes, int n_in,
                              void* d_out, int out_size, void* d_ws, size_t ws_size,
                              hipStream_t stream) {
    const float* x    = (const float*)d_in[0];
    const int*   esrc = (const int*)d_in[1];
    const int*   edst = (const int*)d_in[2];
    const float* w1   = (const float*)d_in[3];
    const float* b1   = (const float*)d_in[4];
    const float* w2   = (const float*)d_in[5];
    const float* b2   = (const float*)d_in[6];
    float* out = (float*)d_out;

    const int N = in_sizes[0] / DIM;   // 50000
    const int E = in_sizes[1];         // 800000

    // workspace layout (floats): out_deg[N] | in_deg[N] | norm[E] | agg[N*64]
    float* ws      = (float*)d_ws;
    float* out_deg = ws;
    float* in_deg  = ws + N;
    float* normv   = ws + 2 * (size_t)N;
    float* agg     = ws + 2 * (size_t)N + E;

    const int totalZero = 2 * N + E + N * DIM;   // zero everything (norm overwritten anyway)
    gcn_zero_kernel<<<4096, 256, 0, stream>>>(ws, totalZero);

    gcn_degree_kernel<<<(E + 255) / 256, 256, 0, stream>>>(esrc, edst, out_deg, in_deg, E);
    gcn_norm_kernel<<<(E + 255) / 256, 256, 0, stream>>>(esrc, edst, out_deg, in_deg, normv, E);

    const int scatterThreads = E * 16;           // 12.8M
    gcn_scatter_kernel<<<(scatterThreads + 255) / 256, 256, 0, stream>>>(
        x, esrc, edst, normv, agg, E);

    const int nTiles = (N + 15) / 16;            // 3125 (exact)
    gcn_ffn_wmma_kernel<<<(nTiles + 3) / 4, 128, 0, stream>>>(
        agg, w1, b1, w2, b2, out, nTiles);
}